// Resampler_35794257445218
// MI455X (gfx1250) — compile-verified
//
#include <hip/hip_runtime.h>
#include <math.h>

// ---------------------------------------------------------------------------
// Kaldi LinearResample 44100 -> 16000, lpfw=6, on [16, 2646000] f32.
//   base=gcd=100, in_unit=441, out_unit=160, Wwin=34, min_first=-16, Wtot=472
//   blocks B=6000, tot_out=960000 per channel (exactly 6000*160, no tail)
// GEMM view per channel:  D[160,6000] = W_emb[160,472] @ X[472,6000],
//   X[k,b] = x_padded[b*441 + k].  A 16-phase row tile only touches a
//   <=76-wide K window, so each wave runs K=80 (20 x WMMA f32 16x16x4).
// Memory-bound: ~230MB traffic -> ~10us at 23.3 TB/s; input staged to LDS
// once per (channel, 16-block) tile and reused by all 10 waves / 160 phases.
// ---------------------------------------------------------------------------

typedef __attribute__((ext_vector_type(2))) float v2f;
typedef __attribute__((ext_vector_type(4))) float v4f;
typedef __attribute__((ext_vector_type(8))) float v8f;

#define ORIGF   44100
#define NEWF    16000
#define IN_UNIT 441
#define OUT_UNIT 160
#define LWAVE   2646000
#define NBLK    6000            // output polyphase blocks per channel
#define TOTOUT  960000          // output samples per channel
#define MTILES  10              // 160 phases / 16
#define KW      80              // per-tile K window (>= 76), multiple of 4
#define NXLDS   7104            // staged floats: 15*441 + max(k0)+KW = 7092, padded
#define PI_D    3.14159265358979323846

__device__ __forceinline__ int k0_of_tile(int mt) {
    // first[i] = ceil((i/new - ww)*orig);  k0 = first[16*mt] - min_first
    const double cutoff = 0.99 * 0.5 * (double)NEWF;          // 7920
    const double ww     = 6.0 / (2.0 * cutoff);
    double t = (double)(mt * 16) / (double)NEWF;
    return (int)ceil((t - ww) * (double)ORIGF) + 16;
}

// Build compact per-tile filter Wc[mt][m][k] (k fastest), 10*16*80 floats.
// Same fp64 expressions as the numpy reference; entries outside a phase's
// +-ww window evaluate to exactly 0, matching the embedded-zero layout.
__global__ void rs_init_filter(float* __restrict__ Wc) {
    const double orig = (double)ORIGF, nf = (double)NEWF, lpfw = 6.0;
    const double cutoff = 0.99 * 0.5 * nf;
    const double ww = lpfw / (2.0 * cutoff);
    int tid = blockIdx.x * blockDim.x + threadIdx.x;
    for (int e = tid; e < MTILES * 16 * KW; e += gridDim.x * blockDim.x) {
        int k  = e % KW;
        int m  = (e / KW) % 16;
        int mt = e / (KW * 16);
        int i  = mt * 16 + m;                     // phase
        double t = (double)i / nf;
        int idx = k0_of_tile(mt) + k - 16;        // absolute input sample index
        double dt = (double)idx / orig - t;
        double w = 0.0;
        if (fabs(dt) < ww) {
            w = 0.5 * (1.0 + cos(2.0 * PI_D * cutoff / lpfw * dt));
            w *= (dt == 0.0) ? (2.0 * cutoff)
                             : (sin(2.0 * PI_D * cutoff * dt) / (PI_D * dt));
            w /= orig;
        }
        Wc[e] = (float)w;
    }
}

// One workgroup = 320 threads = 10 wave32s; grid = (375 block-tiles, 16 ch).
// Wave w computes the 16x16 tile: phases 16w..16w+15  x  blocks b0..b0+15.
__global__ void __launch_bounds__(320)
rs_wmma(const float* __restrict__ wav, const float* __restrict__ Wc,
        float* __restrict__ out) {
    __shared__ float xs[NXLDS];

    const int ntile = blockIdx.x;            // 0..374
    const int c     = blockIdx.y;            // 0..15
    const int b0    = ntile * 16;
    const long s    = (long)b0 * IN_UNIT - 16;   // waveform index of xs[0]
    const float* wrow = wav + (size_t)c * LWAVE;

    // Stage input window (coalesced); zero-pad both edges like the reference.
    for (int p = threadIdx.x; p < NXLDS; p += 320) {
        long g = s + p;
        xs[p] = (g >= 0 && g < LWAVE) ? wrow[g] : 0.0f;
    }
    __syncthreads();

    const int wave = threadIdx.x >> 5;       // = m-tile, 0..9
    const int lane = threadIdx.x & 31;
    const int mn   = lane & 15;              // A: row m, B: col n (block)
    const int kh   = lane >> 4;              // K half-select per ISA layout

    const int k0 = k0_of_tile(wave);
    // A: lane holds W[16*wave+m][kb+2kh .. +1]  (8B-aligned, L2-resident)
    const float* Ap = Wc + ((size_t)(wave * 16 + mn) * KW + 2 * kh);
    // B: lane holds X[kb+2kh .. +1][n] = xs[n*441 + k0 + kb + 2kh ...]
    const float* Bp = xs + mn * IN_UNIT + k0 + 2 * kh;

    v8f acc = {};
#pragma unroll
    for (int kb = 0; kb < KW; kb += 4) {
        v2f a = *(const v2f*)(Ap + kb);
        v2f b; b.x = Bp[kb]; b.y = Bp[kb + 1];
        // D = A(16x4) * B(4x16) + C, full f32
        acc = __builtin_amdgcn_wmma_f32_16x16x4_f32(
            /*neg_a=*/false, a, /*neg_b=*/false, b,
            /*c_mod=*/(short)0, acc, /*reuse_a=*/false, /*reuse_b=*/false);
    }

    // D layout: VGPR r -> M = r + 8*kh, N = mn. out[c][(b0+n)*160 + 16*wave + M]
    // => each lane's 8 results are CONTIGUOUS floats: two b128 stores.
    float* op = out + (size_t)c * TOTOUT
                    + (size_t)(b0 + mn) * OUT_UNIT + wave * 16 + kh * 8;
    v4f lo = {acc[0], acc[1], acc[2], acc[3]};
    v4f hi = {acc[4], acc[5], acc[6], acc[7]};
    *(v4f*)op       = lo;
    *((v4f*)op + 1) = hi;
}

extern "C" void kernel_launch(void* const* d_in, const int* in_sizes, int n_in,
                              void* d_out, int out_size, void* d_ws, size_t ws_size,
                              hipStream_t stream) {
    (void)in_sizes; (void)n_in; (void)out_size; (void)ws_size;
    const float* wav = (const float*)d_in[0];   // [16, 2646000] f32
    float* Wc  = (float*)d_ws;                  // 10*16*80 floats = 51.2 KB
    float* out = (float*)d_out;                 // [16, 960000] f32

    rs_init_filter<<<dim3(50), dim3(256), 0, stream>>>(Wc);
    rs_wmma<<<dim3(NBLK / 16, 16), dim3(320), 0, stream>>>(wav, Wc, out);
}